// FrequencyAwareSumAttention_3951369912995
// MI455X (gfx1250) — compile-verified
//
#include <hip/hip_runtime.h>
#include <hip/hip_bf16.h>

// ---------------------------------------------------------------------------
// FrequencyAwareSumAttention for MI455X (gfx1250), fp32 via V_WMMA_F32_16X16X4_F32.
// Bandwidth-bound op (~214MB min traffic @ 23.3TB/s ~ 9us; only ~50 GFLOP), so
// keep fp32 precision and fuse everything per (b,h,j) inside the 320KB WGP LDS.
// All LDS tiles use even strides so A/B fragments are contiguous 8B pairs
// (single ds_load_b64, no shuffle moves) and remain bank-conflict free.
// Staging in kernel 2 uses GLOBAL_LOAD_ASYNC_TO_LDS_B32 (ASYNCcnt path).
// ---------------------------------------------------------------------------

typedef __attribute__((ext_vector_type(2))) float v2f;
typedef __attribute__((ext_vector_type(8))) float v8f;

#define NB 4
#define NT 256
#define NJ 17
#define NC 512
#define NH 8
#define ND 64
#define NM (NB * NT * NJ) /* 17408 */

#if defined(__has_builtin)
#if __has_builtin(__builtin_amdgcn_global_load_async_to_lds_b32) && \
    __has_builtin(__builtin_amdgcn_s_wait_asynccnt)
#define USE_ASYNC_LDS 1
#endif
#endif

#ifdef USE_ASYNC_LDS
typedef __attribute__((address_space(1))) int gas_int;  // global (__device__) int*
typedef __attribute__((address_space(3))) int las_int;  // LDS (__shared__) int*
#endif

__device__ __forceinline__ v8f wmma4(v2f a, v2f b, v8f c) {
  // D(16x16,f32) = A(16x4,f32) * B(4x16,f32) + C
  return __builtin_amdgcn_wmma_f32_16x16x4_f32(
      /*neg_a=*/false, a, /*neg_b=*/false, b,
      /*c_mod=*/(short)0, c, /*reuse_a=*/false, /*reuse_b=*/false);
}

__device__ __forceinline__ v2f lds2(const float* p) { return *(const v2f*)p; }

// ---------------------------------------------------------------------------
// Kernel 1: qkv = x @ w_qkv   (M=17408, K=512, N=1536), scatter to q/k/v in
// (B,H,J,T,d) layout. Block = 256 thr = 8 waves, 64(M) x 32(N) tile.
// A tile [64][k:34], B tile stored transposed [n=32][k:34].
// ---------------------------------------------------------------------------
__global__ __launch_bounds__(256) void qkv_gemm_kernel(
    const float* __restrict__ x, const float* __restrict__ w_qkv,
    float* __restrict__ qbuf, float* __restrict__ kbuf, float* __restrict__ vbuf) {
  __shared__ __align__(16) float As[64 * 34];
  __shared__ __align__(16) float Bt[32 * 34];
  const int tid = threadIdx.x;
  const int lane = tid & 31;
  const int wv = tid >> 5;
  const int lrow = lane & 15;
  const int lhi = lane >> 4;
  const int wm = wv & 3;   // M sub-tile 0..3
  const int wn = wv >> 2;  // N sub-tile 0..1
  const int mb = blockIdx.x * 64;
  const int nb = blockIdx.y * 32;

  v8f acc = {};
  for (int k0 = 0; k0 < 512; k0 += 32) {
#pragma unroll
    for (int i = 0; i < 8; ++i) {
      int e = tid + i * 256;
      int r = e >> 5, c = e & 31;
      As[r * 34 + c] = x[(size_t)(mb + r) * 512 + k0 + c];
    }
#pragma unroll
    for (int i = 0; i < 4; ++i) {
      int e = tid + i * 256;
      int r = e >> 5, c = e & 31;  // r = k, c = n  (store transposed)
      Bt[c * 34 + r] = w_qkv[(size_t)(k0 + r) * 1536 + nb + c];
    }
    // prefetch next A tile into cache hierarchy (global_prefetch_b8)
    if (k0 + 32 < 512) __builtin_prefetch(&x[(size_t)(mb + (tid >> 5)) * 512 + k0 + 32], 0, 3);
    __syncthreads();
#pragma unroll
    for (int kk = 0; kk < 32; kk += 4) {
      v2f a = lds2(&As[(wm * 16 + lrow) * 34 + kk + 2 * lhi]);
      v2f b = lds2(&Bt[(wn * 16 + lrow) * 34 + kk + 2 * lhi]);
      acc = wmma4(a, b, acc);
    }
    __syncthreads();
  }
  // scatter D: m -> (b,t,j), n -> (which,h,dd); dst (B,H,J,T,d)
#pragma unroll
  for (int r = 0; r < 8; ++r) {
    int m = mb + wm * 16 + r + 8 * lhi;
    int n = nb + wn * 16 + lrow;
    int j = m % 17;
    int tmp = m / 17;
    int t = tmp & 255;
    int b = tmp >> 8;
    int which = n >> 9;
    int c = n & 511;
    int h = c >> 6;
    int dd = c & 63;
    size_t dst = ((((size_t)b * 8 + h) * 17 + j) * 256 + t) * 64 + dd;
    float val = acc[r];
    if (which == 0) qbuf[dst] = val;
    else if (which == 1) kbuf[dst] = val;
    else vbuf[dst] = val;
  }
}

// ---------------------------------------------------------------------------
// Kernel 2: fused per-(b,h,j) attention. One block per (b,h,j).
//   LDS: qs/ks [256][68], vt (v transposed) [64][260],
//        union{ qf/kf re+im 4x[64][68] | scores [64][260] }, af [64][68].
//   = 73216 floats = 286 KB dynamic LDS (fits 320KB WGP LDS).
// rfft realized as DFT-GEMM with cos/sin generated in registers.
// ---------------------------------------------------------------------------
#define LQ 68
#define LV 260
#define LS 260
#define LF 68

__global__ __launch_bounds__(256) void attn_fused_kernel(
    const float* __restrict__ qbuf, const float* __restrict__ kbuf,
    const float* __restrict__ vbuf, const float* __restrict__ att_map,
    const float* __restrict__ weight, const float* __restrict__ freq_gate,
    float* __restrict__ ctxbuf) {
  extern __shared__ float smem[];
  float* qs = smem;                  // 256*68
  float* ks = qs + 256 * LQ;         // 256*68
  float* vt = ks + 256 * LQ;         // 64*260  (v transposed: [dd][t])
  float* un = vt + 64 * LV;          // union: 4*64*68=17408 | 64*260=16640
  float* qfr = un;
  float* qfi = un + 64 * LF;
  float* kfr = un + 2 * 64 * LF;
  float* kfi = un + 3 * 64 * LF;
  float* st = un;                    // scores 64 x 256, stride 260
  float* af = un + 4 * 64 * LF;      // 64*68 spectral attention

  const int tid = threadIdx.x;
  const int lane = tid & 31;
  const int wv = tid >> 5;
  const int lrow = lane & 15;
  const int lhi = lane >> 4;
  const int bhj = blockIdx.x;
  const int j = bhj % 17;
  const int h = (bhj / 17) % 8;
  const int b = bhj / (17 * 8);

  const float scale = 0.125f;  // 64^-0.5
  const float wgt = weight[0];
  const float gate = 1.0f / (1.0f + __expf(-freq_gate[0]));

  const float* qg = qbuf + (size_t)bhj * 256 * 64;
  const float* kg = kbuf + (size_t)bhj * 256 * 64;
  const float* vg = vbuf + (size_t)bhj * 256 * 64;
  const float* ag = att_map + (size_t)bhj * 256 * 256;

  // stage q,k (row-major) and v (transposed) into LDS
#ifdef USE_ASYNC_LDS
  for (int i = 0; i < 64; ++i) {
    int e = tid + i * 256;
    int r = e >> 6, c = e & 63;
    __builtin_amdgcn_global_load_async_to_lds_b32(
        (gas_int*)(unsigned long long)(qg + e),
        (las_int*)(unsigned int)(unsigned long long)(&qs[r * LQ + c]), 0, 0);
    __builtin_amdgcn_global_load_async_to_lds_b32(
        (gas_int*)(unsigned long long)(kg + e),
        (las_int*)(unsigned int)(unsigned long long)(&ks[r * LQ + c]), 0, 0);
    __builtin_amdgcn_global_load_async_to_lds_b32(
        (gas_int*)(unsigned long long)(vg + e),
        (las_int*)(unsigned int)(unsigned long long)(&vt[c * LV + r]), 0, 0);
  }
  __builtin_amdgcn_s_wait_asynccnt(0);
#else
  for (int i = 0; i < 64; ++i) {
    int e = tid + i * 256;
    int r = e >> 6, c = e & 63;
    qs[r * LQ + c] = qg[e];
    ks[r * LQ + c] = kg[e];
    vt[c * LV + r] = vg[e];
  }
#endif
  __syncthreads();

  // ---- Phase A: rfft over T as DFT GEMM: (64f x 256t) @ (256t x 64d) ----
  // X[f] = sum_t x[t] e^{-i*2pi*f*t/256}; cos/sin generated in registers.
  const float W0 = -6.283185307179586f / 256.0f;
#pragma unroll
  for (int pi = 0; pi < 2; ++pi) {
    int p = wv + pi * 8;  // 16 tile positions over 8 waves
    int fb = (p >> 2) * 16;
    int db = (p & 3) * 16;
    int f = fb + lrow;
    v8f qre = {}, qim = {}, kre = {}, kim = {};
    for (int kk = 0; kk < 256; kk += 4) {
      int t0 = kk + 2 * lhi;
      float a0 = W0 * (float)(f * t0);
      float a1 = W0 * (float)(f * (t0 + 1));
      v2f ca, sa, bq, bk;
      ca.x = __cosf(a0); ca.y = __cosf(a1);
      sa.x = __sinf(a0); sa.y = __sinf(a1);
      bq.x = qs[(t0 + 0) * LQ + db + lrow];
      bq.y = qs[(t0 + 1) * LQ + db + lrow];
      bk.x = ks[(t0 + 0) * LQ + db + lrow];
      bk.y = ks[(t0 + 1) * LQ + db + lrow];
      qre = wmma4(ca, bq, qre);
      qim = wmma4(sa, bq, qim);
      kre = wmma4(ca, bk, kre);
      kim = wmma4(sa, bk, kim);
    }
#pragma unroll
    for (int r = 0; r < 8; ++r) {
      int fo = fb + r + 8 * lhi;
      int dc = db + lrow;
      qfr[fo * LF + dc] = qre[r];
      qfi[fo * LF + dc] = qim[r];
      kfr[fo * LF + dc] = kre[r];
      kfi[fo * LF + dc] = kim[r];
    }
  }
  __syncthreads();

  // ---- Phase B: af = scale*(qre@kre^T + qim@kim^T); softmax rows ----
#pragma unroll
  for (int pi = 0; pi < 2; ++pi) {
    int p = wv + pi * 8;
    int fb = (p >> 2) * 16;
    int gb = (p & 3) * 16;
    v8f acc = {};
    for (int kk = 0; kk < 64; kk += 4) {
      int k0 = kk + 2 * lhi;
      acc = wmma4(lds2(&qfr[(fb + lrow) * LF + k0]),
                  lds2(&kfr[(gb + lrow) * LF + k0]), acc);
      acc = wmma4(lds2(&qfi[(fb + lrow) * LF + k0]),
                  lds2(&kfi[(gb + lrow) * LF + k0]), acc);
    }
#pragma unroll
    for (int r = 0; r < 8; ++r)
      af[(fb + r + 8 * lhi) * LF + gb + lrow] = acc[r] * scale;
  }
  __syncthreads();
  if (tid < 64) {  // softmax over g (64 elems) per f row
    float mx = -1e30f;
    for (int g = 0; g < 64; ++g) mx = fmaxf(mx, af[tid * LF + g]);
    float sum = 0.f;
    for (int g = 0; g < 64; ++g) {
      float e = __expf(af[tid * LF + g] - mx);
      af[tid * LF + g] = e;
      sum += e;
    }
    float inv = 1.0f / sum;
    for (int g = 0; g < 64; ++g) af[tid * LF + g] *= inv;
  }
  __syncthreads();

  // ---- Phase C: per 64-row t-tile: scores, softmax, blend, @v ----
  for (int t0 = 0; t0 < 256; t0 += 64) {
    // C1: raw scores (64 x 256) = q[t0:t0+64] @ k^T, scaled
#pragma unroll
    for (int pi = 0; pi < 8; ++pi) {
      int p = wv * 8 + pi;  // 64 tiles over 8 waves
      int tm = p >> 4, tn = p & 15;
      v8f acc = {};
      for (int kk = 0; kk < 64; kk += 4) {
        int k0 = kk + 2 * lhi;
        acc = wmma4(lds2(&qs[(t0 + tm * 16 + lrow) * LQ + k0]),
                    lds2(&ks[(tn * 16 + lrow) * LQ + k0]), acc);
      }
#pragma unroll
      for (int r = 0; r < 8; ++r)
        st[(tm * 16 + r + 8 * lhi) * LS + tn * 16 + lrow] = acc[r] * scale;
    }
    __syncthreads();
    // C2: softmax over s (256) per t row
    if (tid < 64) {
      float mx = -1e30f;
      for (int s = 0; s < 256; ++s) mx = fmaxf(mx, st[tid * LS + s]);
      float sum = 0.f;
      for (int s = 0; s < 256; ++s) {
        float e = __expf(st[tid * LS + s] - mx);
        st[tid * LS + s] = e;
        sum += e;
      }
      float inv = 1.0f / sum;
      for (int s = 0; s < 256; ++s) st[tid * LS + s] *= inv;
    }
    __syncthreads();
    // C3: blend: bilinear-upsampled spectral attn + time attn + att_map
    for (int e = tid; e < 64 * 256; e += 256) {
      int tt = e >> 8, s = e & 255;
      int t = t0 + tt;
      float srcf = fmaxf((t + 0.5f) * (129.0f / 256.0f) - 0.5f, 0.0f);
      int i0 = (int)srcf;
      int i1 = min(i0 + 1, 128);
      float wf = srcf - (float)i0;
      float srcg = fmaxf((s + 0.5f) * (64.0f / 256.0f) - 0.5f, 0.0f);
      int g0 = (int)srcg;
      int g1 = min(g0 + 1, 63);
      float wg = srcg - (float)g0;
      float v00 = (i0 < 64) ? af[i0 * LF + g0] : 0.f;  // rows >=64 are zero-pad
      float v01 = (i0 < 64) ? af[i0 * LF + g1] : 0.f;
      float v10 = (i1 < 64) ? af[i1 * LF + g0] : 0.f;
      float v11 = (i1 < 64) ? af[i1 * LF + g1] : 0.f;
      float fr = (v00 * (1.f - wg) + v01 * wg) * (1.f - wf) +
                 (v10 * (1.f - wg) + v11 * wg) * wf;
      float at = st[tt * LS + s];
      float am = ag[(size_t)t * 256 + s];
      st[tt * LS + s] = wgt * (gate * fr + (1.f - gate) * at) + (1.f - wgt) * am;
    }
    __syncthreads();
    // C4: ctx (64 x 64) = attn(64x256) @ v(256x64); store (B,T,J,H*d)
#pragma unroll
    for (int pi = 0; pi < 2; ++pi) {
      int p = wv + pi * 8;
      int tm = p >> 2, tn = p & 3;
      v8f acc = {};
      for (int kk = 0; kk < 256; kk += 4) {
        int k0 = kk + 2 * lhi;
        acc = wmma4(lds2(&st[(tm * 16 + lrow) * LS + k0]),
                    lds2(&vt[(tn * 16 + lrow) * LV + k0]), acc);
      }
#pragma unroll
      for (int r = 0; r < 8; ++r) {
        int t = t0 + tm * 16 + r + 8 * lhi;
        int dd = tn * 16 + lrow;
        ctxbuf[(((size_t)b * 256 + t) * 17 + j) * 512 + h * 64 + dd] = acc[r];
      }
    }
    __syncthreads();
  }
}

// ---------------------------------------------------------------------------
// Kernel 3: out = ctx @ w_proj + b_proj  (M=17408, K=512, N=512)
// ---------------------------------------------------------------------------
__global__ __launch_bounds__(256) void proj_gemm_kernel(
    const float* __restrict__ ctx, const float* __restrict__ w_proj,
    const float* __restrict__ b_proj, float* __restrict__ out) {
  __shared__ __align__(16) float As[64 * 34];
  __shared__ __align__(16) float Bt[32 * 34];
  const int tid = threadIdx.x;
  const int lane = tid & 31;
  const int wv = tid >> 5;
  const int lrow = lane & 15;
  const int lhi = lane >> 4;
  const int wm = wv & 3;
  const int wn = wv >> 2;
  const int mb = blockIdx.x * 64;
  const int nb = blockIdx.y * 32;

  v8f acc = {};
  for (int k0 = 0; k0 < 512; k0 += 32) {
#pragma unroll
    for (int i = 0; i < 8; ++i) {
      int e = tid + i * 256;
      int r = e >> 5, c = e & 31;
      As[r * 34 + c] = ctx[(size_t)(mb + r) * 512 + k0 + c];
    }
#pragma unroll
    for (int i = 0; i < 4; ++i) {
      int e = tid + i * 256;
      int r = e >> 5, c = e & 31;
      Bt[c * 34 + r] = w_proj[(size_t)(k0 + r) * 512 + nb + c];
    }
    __syncthreads();
#pragma unroll
    for (int kk = 0; kk < 32; kk += 4) {
      v2f a = lds2(&As[(wm * 16 + lrow) * 34 + kk + 2 * lhi]);
      v2f b = lds2(&Bt[(wn * 16 + lrow) * 34 + kk + 2 * lhi]);
      acc = wmma4(a, b, acc);
    }
    __syncthreads();
  }
#pragma unroll
  for (int r = 0; r < 8; ++r) {
    int m = mb + wm * 16 + r + 8 * lhi;
    int n = nb + wn * 16 + lrow;
    out[(size_t)m * 512 + n] = acc[r] + b_proj[n];
  }
}

// ---------------------------------------------------------------------------
extern "C" void kernel_launch(void* const* d_in, const int* in_sizes, int n_in,
                              void* d_out, int out_size, void* d_ws, size_t ws_size,
                              hipStream_t stream) {
  (void)in_sizes; (void)n_in; (void)out_size; (void)ws_size;
  const float* x = (const float*)d_in[0];
  const float* att_map = (const float*)d_in[1];
  const float* weight = (const float*)d_in[2];
  const float* w_qkv = (const float*)d_in[3];
  const float* w_proj = (const float*)d_in[4];
  const float* b_proj = (const float*)d_in[5];
  const float* freq_gate = (const float*)d_in[6];
  float* out = (float*)d_out;

  // workspace: q,k,v,ctx each B*H*J*T*d = 8,912,896 f32 (total ~143 MB)
  const size_t SZ = (size_t)NB * NH * NJ * NT * ND;
  float* ws = (float*)d_ws;
  float* qbuf = ws;
  float* kbuf = ws + SZ;
  float* vbuf = ws + 2 * SZ;
  float* ctxb = ws + 3 * SZ;

  qkv_gemm_kernel<<<dim3(NM / 64, 1536 / 32), 256, 0, stream>>>(
      x, w_qkv, qbuf, kbuf, vbuf);

  // dyn LDS: 2*(256*68) + 64*260 + max(4*64*68, 64*260) + 64*68 = 73216 floats
  const size_t smem2 = (size_t)(2 * 256 * LQ + 64 * LV + 4 * 64 * LF + 64 * LF) * sizeof(float);
  attn_fused_kernel<<<dim3(NB * NH * NJ), 256, smem2, stream>>>(
      qbuf, kbuf, vbuf, att_map, weight, freq_gate, ctxb);

  proj_gemm_kernel<<<dim3(NM / 64, 512 / 32), 256, 0, stream>>>(
      ctxb, w_proj, b_proj, out);
}